// ThreeDGNNModule_69174743269615
// MI455X (gfx1250) — compile-verified
//
#include <hip/hip_runtime.h>

// ---------------------------------------------------------------------------
// 3D-GNN: dist/top-k once, then 3x { y=relu(mlp(h)); msg=mean_k y[idx];
//                                    h=relu(rnn([h,msg])) }
// GEMMs run on v_wmma_f32_16x16x4_f32 (fp32 WMMA, reference precision).
// Each wave computes a 16x64 output tile (4 accumulators, A-fragment reuse).
// ---------------------------------------------------------------------------

typedef float v2f __attribute__((ext_vector_type(2)));
typedef float v8f __attribute__((ext_vector_type(8)));

#define NB   2
#define CC   256
#define HWD  3600       // 60*60
#define KNN  64
#define RTOT (NB * HWD) // 7200 rows total (divisible by 16)

// ---------------------------------------------------------------------------
// Top-K=64 nearest neighbors per point. One block per (n,s).
// Key = (float_bits(dist) << 32) | index : dist >= 0 so IEEE bits are
// monotone as unsigned; u64-min gives smallest distance with lowest-index
// tie-break (matches jax.lax.top_k stability). Two barriers per round.
// ---------------------------------------------------------------------------
__global__ void topk_kernel(const float* __restrict__ pts, int* __restrict__ idx) {
    __shared__ float dist[HWD];
    __shared__ unsigned long long wmin[8];
    const int b    = blockIdx.x;          // 0 .. RTOT-1
    const int n    = b / HWD;
    const int s    = b % HWD;
    const int tid  = threadIdx.x;
    const int lane = tid & 31;
    const int wid  = tid >> 5;
    const float* p = pts + (size_t)n * 3 * HWD;   // (3, HW) for this batch
    const float px = p[s], py = p[HWD + s], pz = p[2 * HWD + s];

    for (int j = tid; j < HWD; j += 256) {
        float dx = px - p[j];
        float dy = py - p[HWD + j];
        float dz = pz - p[2 * HWD + j];
        dist[j] = dx * dx + dy * dy + dz * dz;
    }
    __syncthreads();

    for (int k = 0; k < KNN; ++k) {
        unsigned long long best = ~0ull;
        for (int j = tid; j < HWD; j += 256) {
            unsigned long long key =
                ((unsigned long long)__float_as_uint(dist[j]) << 32) | (unsigned)j;
            best = key < best ? key : best;
        }
        // wave32 u64-min reduction
        #pragma unroll
        for (int off = 16; off > 0; off >>= 1) {
            unsigned long long o = __shfl_xor(best, off);
            best = o < best ? o : best;
        }
        if (lane == 0) wmin[wid] = best;
        __syncthreads();
        if (tid == 0) {
            unsigned long long m = wmin[0];
            #pragma unroll
            for (int w = 1; w < 8; ++w) m = wmin[w] < m ? wmin[w] : m;
            int bi = (int)(unsigned)(m & 0xffffffffu);
            idx[(size_t)b * KNN + k] = bi;
            dist[bi] = 3.4e38f;   // remove from candidate set
        }
        __syncthreads();
    }
}

// ---------------------------------------------------------------------------
// (N,C,HW) -> row-major (N*HW, C)
// ---------------------------------------------------------------------------
__global__ void nchw_to_rows(const float* __restrict__ src, float* __restrict__ dst) {
    int t = blockIdx.x * 256 + threadIdx.x;
    if (t >= RTOT * CC) return;
    int c   = t % CC;
    int row = t / CC;
    int s   = row % HWD;
    int n   = row / HWD;
    dst[t] = src[((size_t)n * CC + c) * HWD + s];
}

// ---------------------------------------------------------------------------
// C[m][n] = relu( sum_k A[m][k] * W[n][k] + bias[n] )
// A: (RTOT x Kdim) row-major, W: (256 x Kdim) row-major (torch Linear weight),
// C: (RTOT x 256) row-major. One 16x64 tile per wave32 (4 accumulators,
// A fragment reused across 4 WMMAs), fp32 WMMA, K-step 4.
//
// Fragment layouts (CDNA5 ISA 7.12.2, 32-bit):
//   A 16x4 : lane l -> row m=l&15; VGPR0/1 hold K = 2*(l>>4) + {0,1}
//   B 4x16 : lane l -> col n=l&15; VGPR0/1 hold K = 2*(l>>4) + {0,1}
//   C 16x16: VGPR v -> row = v + 8*(l>>4), col = l&15
// ---------------------------------------------------------------------------
__global__ void wmma_gemm_relu(const float* __restrict__ A,
                               const float* __restrict__ W,
                               const float* __restrict__ bias,
                               float* __restrict__ C,
                               int Kdim) {
    const int wave = (blockIdx.x * blockDim.x + threadIdx.x) >> 5;
    const int lane = threadIdx.x & 31;
    const int m0   = (wave >> 2) << 4;   // 4 n-groups of width 64 (256/64)
    const int n0   = (wave & 3) << 6;
    const int lo   = lane & 15;
    const int hi   = lane >> 4;

    const float* arow = A + (size_t)(m0 + lo) * Kdim + (hi << 1);
    const float* b0   = W + (size_t)(n0 + 0  + lo) * Kdim + (hi << 1);
    const float* b1   = W + (size_t)(n0 + 16 + lo) * Kdim + (hi << 1);
    const float* b2   = W + (size_t)(n0 + 32 + lo) * Kdim + (hi << 1);
    const float* b3   = W + (size_t)(n0 + 48 + lo) * Kdim + (hi << 1);

    v8f acc0 = {}, acc1 = {}, acc2 = {}, acc3 = {};
    for (int k0 = 0; k0 < Kdim; k0 += 4) {
        v2f a  = *(const v2f*)(arow + k0);
        v2f w0 = *(const v2f*)(b0 + k0);
        v2f w1 = *(const v2f*)(b1 + k0);
        v2f w2 = *(const v2f*)(b2 + k0);
        v2f w3 = *(const v2f*)(b3 + k0);
        acc0 = __builtin_amdgcn_wmma_f32_16x16x4_f32(false, a, false, w0, (short)0, acc0, false, false);
        acc1 = __builtin_amdgcn_wmma_f32_16x16x4_f32(false, a, false, w1, (short)0, acc1, false, false);
        acc2 = __builtin_amdgcn_wmma_f32_16x16x4_f32(false, a, false, w2, (short)0, acc2, false, false);
        acc3 = __builtin_amdgcn_wmma_f32_16x16x4_f32(false, a, false, w3, (short)0, acc3, false, false);
    }

    float* crow = C + (size_t)(m0 + (hi << 3)) * CC + n0 + lo;
    const v8f accs[4] = {acc0, acc1, acc2, acc3};
#pragma unroll
    for (int j = 0; j < 4; ++j) {
        const float bv = bias[n0 + (j << 4) + lo];
#pragma unroll
        for (int v = 0; v < 8; ++v) {
            float x = accs[j][v] + bv;
            crow[(size_t)v * CC + (j << 4)] = x > 0.0f ? x : 0.0f;
        }
    }
}

// ---------------------------------------------------------------------------
// nh[row] = [ h[row] , mean_k y[n, idx[row,k]] ]   (row-major, 512 wide)
// Block per row, thread per channel; idx staged in LDS; y reads coalesced
// (1KB full row per k, L2-resident).
// ---------------------------------------------------------------------------
__global__ void gather_mean_concat(const float* __restrict__ h,
                                   const float* __restrict__ y,
                                   const int* __restrict__ idx,
                                   float* __restrict__ nh) {
    __shared__ int nbr[KNN];
    const int row = blockIdx.x;
    const int c   = threadIdx.x;
    const int n   = row / HWD;
    if (c < KNN) nbr[c] = idx[(size_t)row * KNN + c];
    __syncthreads();

    nh[(size_t)row * (2 * CC) + c] = h[(size_t)row * CC + c];

    const float* yb = y + (size_t)n * HWD * CC;
    float acc = 0.0f;
#pragma unroll 8
    for (int k = 0; k < KNN; ++k)
        acc += yb[(size_t)nbr[k] * CC + c];
    nh[(size_t)row * (2 * CC) + CC + c] = acc * (1.0f / KNN);
}

// ---------------------------------------------------------------------------
// out (N, 2C, HW): channels [0,256) = cnn_feature, [256,512) = h (rows->NCHW)
// ---------------------------------------------------------------------------
__global__ void write_out(const float* __restrict__ cnn,
                          const float* __restrict__ h,
                          float* __restrict__ out) {
    int t = blockIdx.x * 256 + threadIdx.x;
    if (t >= NB * 2 * CC * HWD) return;
    int s  = t % HWD;
    int ch = (t / HWD) % (2 * CC);
    int n  = t / (HWD * 2 * CC);
    float v;
    if (ch < CC) v = cnn[((size_t)n * CC + ch) * HWD + s];
    else         v = h[((size_t)n * HWD + s) * CC + (ch - CC)];
    out[t] = v;
}

// ---------------------------------------------------------------------------
extern "C" void kernel_launch(void* const* d_in, const int* in_sizes, int n_in,
                              void* d_out, int out_size, void* d_ws, size_t ws_size,
                              hipStream_t stream) {
    const float* cnn   = (const float*)d_in[0];  // (2,256,60,60)
    const float* pts   = (const float*)d_in[1];  // (2,3,60,60)
    const float* mlp_w = (const float*)d_in[2];  // (256,256)
    const float* mlp_b = (const float*)d_in[3];  // (256,)
    const float* rnn_w = (const float*)d_in[4];  // (256,512)
    const float* rnn_b = (const float*)d_in[5];  // (256,)
    float* out = (float*)d_out;

    // Workspace carve-out (~31.4 MB total)
    char* ws = (char*)d_ws;
    int*   idx = (int*)ws;    ws += (size_t)RTOT * KNN * sizeof(int);   // 1.84 MB
    float* h   = (float*)ws;  ws += (size_t)RTOT * CC * sizeof(float);  // 7.37 MB
    float* y   = (float*)ws;  ws += (size_t)RTOT * CC * sizeof(float);  // 7.37 MB
    float* nh  = (float*)ws;                                            // 14.75 MB

    nchw_to_rows<<<(RTOT * CC + 255) / 256, 256, 0, stream>>>(cnn, h);
    topk_kernel<<<RTOT, 256, 0, stream>>>(pts, idx);

    // (RTOT/16) m-tiles * 4 n-groups waves total; 8 waves (256 thr) per block
    const int gemm_blocks = (RTOT / 16) * 4 / 8;  // 225
    for (int it = 0; it < 3; ++it) {
        // y = relu(h @ mlp_w^T + mlp_b)   -- MLP hoisted out of the gather
        wmma_gemm_relu<<<gemm_blocks, 256, 0, stream>>>(h, mlp_w, mlp_b, y, CC);
        // nh = [h, mean_k y[idx]]
        gather_mean_concat<<<RTOT, 256, 0, stream>>>(h, y, idx, nh);
        // h = relu(nh @ rnn_w^T + rnn_b)
        wmma_gemm_relu<<<gemm_blocks, 256, 0, stream>>>(nh, rnn_w, rnn_b, h, 2 * CC);
    }

    write_out<<<(NB * 2 * CC * HWD + 255) / 256, 256, 0, stream>>>(cnn, h, out);
}